// ImageWatermarkDetector2_11819749998776
// MI455X (gfx1250) — compile-verified
//
#include <hip/hip_runtime.h>
#include <hip/hip_bf16.h>

typedef __attribute__((ext_vector_type(2))) float v2f;
typedef __attribute__((ext_vector_type(8))) float v8f;

__device__ __forceinline__ v8f wmma4(v2f a, v2f b, v8f c) {
  // D = A(16x4,f32) * B(4x16,f32) + C(16x16,f32)
  return __builtin_amdgcn_wmma_f32_16x16x4_f32(false, a, false, b, (short)0, c, false, false);
}

// ---------------------------------------------------------------------------
// Fused per-image kernel: conv1+BN+ReLU -> conv2+BN+ReLU -> Haar gate -> attn
// One block per image, 256 threads (8 waves). Everything stays in LDS.
// ---------------------------------------------------------------------------
__global__ __launch_bounds__(256) void fused_conv(
    const float* __restrict__ x,
    const float* __restrict__ w1, const float* __restrict__ b1,
    const float* __restrict__ g1, const float* __restrict__ be1,
    const float* __restrict__ m1, const float* __restrict__ v1,
    const float* __restrict__ w2, const float* __restrict__ b2,
    const float* __restrict__ g2, const float* __restrict__ be2,
    const float* __restrict__ m2, const float* __restrict__ v2,
    const float* __restrict__ wa, const float* __restrict__ ba,
    float* __restrict__ fout)
{
  // LDS layout (floats): hs[8192] persistent, overlay region 24960
  //  phase1: xs[3072] w1s[32*76=2432] As1[256*76=19456]
  //  phase2: As2[64*292=18688] featb[4096] attns[64]
  __shared__ __align__(16) float smem[8192 + 24960];  // 132.6 KB
  float* hs    = smem;
  float* ovl   = smem + 8192;
  float* xs    = ovl;
  float* w1s   = ovl + 3072;
  float* As1   = ovl + 5504;
  float* As2   = ovl;
  float* featb = ovl + 18688;
  float* attns = ovl + 22784;

  const int tid  = threadIdx.x;
  const int lane = tid & 31;
  const int wave = tid >> 5;
  const int half = lane >> 4;   // 0: lanes 0-15, 1: lanes 16-31
  const int l15  = lane & 15;
  const int img  = blockIdx.x;

  // ---- stage input image (3x32x32) and conv1 weights (K padded 75->76) ----
  {
    const float4* xg = (const float4*)(x + (size_t)img * 3072);
    float4* xsv = (float4*)xs;
    for (int i = tid; i < 768; i += 256) xsv[i] = xg[i];
    for (int i = tid; i < 32 * 76; i += 256) {
      int n = i / 76, k = i - n * 76;
      w1s[i] = (k < 75) ? w1[n * 75 + k] : 0.0f;
    }
  }
  __syncthreads();

  // ---- im2col for conv1: 256 output pixels x 76 (3ch * 5x5, s2, p2) ----
  for (int i = tid; i < 256 * 76; i += 256) {
    int m = i / 76, k = i - m * 76;
    float val = 0.0f;
    if (k < 75) {
      int c = k / 25, r = k - c * 25;
      int ky = r / 5, kx = r - ky * 5;
      int oy = m >> 4, ox = m & 15;
      int iy = oy * 2 + ky - 2, ix = ox * 2 + kx - 2;
      if (iy >= 0 && iy < 32 && ix >= 0 && ix < 32)
        val = xs[c * 1024 + iy * 32 + ix];
    }
    As1[i] = val;
  }
  __syncthreads();

  // ---- conv1 as WMMA GEMM: M=256 px, N=32 ch, K=76. 32 tiles / 8 waves ----
  for (int j = 0; j < 4; ++j) {
    int t = wave + 8 * j;
    int mt = t >> 1, nt = t & 1;
    v8f acc = {};
    const float* arow = As1 + (mt * 16 + l15) * 76 + half * 2;
    const float* brow = w1s + (nt * 16 + l15) * 76 + half * 2;
#pragma unroll
    for (int kk = 0; kk < 76; kk += 4) {
      v2f av = *(const v2f*)(arow + kk);
      v2f bv = *(const v2f*)(brow + kk);
      acc = wmma4(av, bv, acc);
    }
    int n = nt * 16 + l15;                       // output channel
    float inv = g1[n] * rsqrtf(v1[n] + 1e-5f);   // folded BN (+conv bias)
    float sh  = be1[n] - m1[n] * inv + b1[n] * inv;
#pragma unroll
    for (int r = 0; r < 8; ++r) {
      int m = mt * 16 + r + half * 8;            // pixel index 0..255
      float vv = acc[r] * inv + sh;
      hs[n * 256 + m] = vv > 0.0f ? vv : 0.0f;   // ReLU
    }
  }
  __syncthreads();

  // ---- im2col for conv2: 64 px x 288 (32ch * 3x3, s2, p1), row stride 292 ----
  for (int i = tid; i < 64 * 288; i += 256) {
    int m = i / 288, k = i - m * 288;
    int c = k / 9, r = k - c * 9;
    int ky = r / 3, kx = r - ky * 3;
    int oy = m >> 3, ox = m & 7;
    int iy = oy * 2 + ky - 1, ix = ox * 2 + kx - 1;
    float val = 0.0f;
    if (iy >= 0 && iy < 16 && ix >= 0 && ix < 16)
      val = hs[c * 256 + iy * 16 + ix];
    As2[m * 292 + k] = val;
  }
  __syncthreads();

  // ---- conv2 as WMMA GEMM: M=64, N=64, K=288; B-frags from L2-resident w2 ----
  for (int j = 0; j < 2; ++j) {
    int t = wave + 8 * j;
    int mt = t >> 2, nt = t & 3;
    int n = nt * 16 + l15;                       // output channel 0..63
    v8f acc = {};
    const float* arow = As2 + (mt * 16 + l15) * 292 + half * 2;
    const float* brow = w2 + n * 288 + half * 2;
#pragma unroll 8
    for (int kk = 0; kk < 288; kk += 4) {
      v2f av = *(const v2f*)(arow + kk);
      v2f bv = *(const v2f*)(brow + kk);
      acc = wmma4(av, bv, acc);
    }
    float inv = g2[n] * rsqrtf(v2[n] + 1e-5f);
    float sh  = be2[n] - m2[n] * inv + b2[n] * inv;
#pragma unroll
    for (int r = 0; r < 8; ++r) {
      int m = mt * 16 + r + half * 8;            // pixel 0..63 (8x8)
      float vv = acc[r] * inv + sh;
      featb[n * 64 + m] = vv > 0.0f ? vv : 0.0f;
    }
  }
  __syncthreads();

  // ---- Haar cD gate, in place (disjoint 2x2 blocks) ----
  for (int i = tid; i < 1024; i += 256) {
    int c = i >> 4, blk = i & 15;
    int by = blk >> 2, bx = blk & 3;
    int base = c * 64 + by * 16 + bx * 2;        // (y=2*by, x=2*bx) in 8x8
    float a = featb[base],     b = featb[base + 1];
    float cc = featb[base + 8], d = featb[base + 9];
    float cD = 0.5f * (a - b - cc + d);
    featb[base]     = a * cD;  featb[base + 1] = b * cD;
    featb[base + 8] = cc * cD; featb[base + 9] = d * cD;
  }
  __syncthreads();

  // ---- 1x1-conv sigmoid attention (per pixel over 64 channels) ----
  if (tid < 64) {
    float s = ba[0];
#pragma unroll 8
    for (int c = 0; c < 64; ++c) s += wa[c] * featb[c * 64 + tid];
    attns[tid] = 1.0f / (1.0f + expf(-s));
  }
  __syncthreads();

  // ---- apply attention, write flattened feature vector ----
  float* fo = fout + (size_t)img * 4096;
  for (int i = tid; i < 4096; i += 256)
    fo[i] = featb[i] * attns[i & 63];
}

// ---------------------------------------------------------------------------
// Tiled WMMA GEMM: C[M,N] = act(A[M,K] * W[N,K]^T + bias[N])
// Block tile 64x32, 8 waves (4x2 grid of 16x16 wave tiles), K chunk 32.
// ---------------------------------------------------------------------------
template <int ACT>   // 0 = ReLU, 1 = sigmoid
__global__ __launch_bounds__(256) void gemm_bias_act(
    const float* __restrict__ A, const float* __restrict__ W,
    const float* __restrict__ bias, float* __restrict__ C,
    int M, int N, int K)
{
  __shared__ __align__(16) float As[64 * 36];   // row stride 36 -> no bank clash
  __shared__ __align__(16) float Ws[32 * 36];
  const int tid  = threadIdx.x;
  const int lane = tid & 31, wave = tid >> 5;
  const int half = lane >> 4, l15 = lane & 15;
  const int m0 = blockIdx.y * 64;
  const int n0 = blockIdx.x * 32;
  const int mt = (wave >> 1) * 16;
  const int nt = (wave & 1) * 16;

  v8f acc = {};
  for (int kk = 0; kk < K; kk += 32) {
#pragma unroll
    for (int v = 0; v < 2; ++v) {               // A chunk: 64x32 = 512 float4
      int idx = tid + v * 256;
      int row = idx >> 3, col4 = (idx & 7) * 4;
      float4 t = *(const float4*)(A + (size_t)(m0 + row) * K + kk + col4);
      *(float4*)(As + row * 36 + col4) = t;
    }
    {                                            // W chunk: 32x32 = 256 float4
      int row = tid >> 3, col4 = (tid & 7) * 4;
      float4 t = *(const float4*)(W + (size_t)(n0 + row) * K + kk + col4);
      *(float4*)(Ws + row * 36 + col4) = t;
    }
    __syncthreads();
    const float* arow = As + (mt + l15) * 36 + half * 2;
    const float* brow = Ws + (nt + l15) * 36 + half * 2;
#pragma unroll
    for (int k8 = 0; k8 < 32; k8 += 4) {
      v2f av = *(const v2f*)(arow + k8);
      v2f bv = *(const v2f*)(brow + k8);
      acc = wmma4(av, bv, acc);
    }
    __syncthreads();
  }
  int n = n0 + nt + l15;
  float bsv = bias[n];
#pragma unroll
  for (int r = 0; r < 8; ++r) {
    int m = m0 + mt + r + half * 8;
    float vv = acc[r] + bsv;
    if (ACT == 0) vv = vv > 0.0f ? vv : 0.0f;
    else          vv = 1.0f / (1.0f + expf(-vv));
    C[(size_t)m * N + n] = vv;
  }
}

extern "C" void kernel_launch(void* const* d_in, const int* in_sizes, int n_in,
                              void* d_out, int out_size, void* d_ws, size_t ws_size,
                              hipStream_t stream) {
  const float* x   = (const float*)d_in[0];
  const float* w1  = (const float*)d_in[1];
  const float* b1  = (const float*)d_in[2];
  const float* g1  = (const float*)d_in[3];
  const float* be1 = (const float*)d_in[4];
  const float* m1  = (const float*)d_in[5];
  const float* v1  = (const float*)d_in[6];
  const float* w2  = (const float*)d_in[7];
  const float* b2  = (const float*)d_in[8];
  const float* g2  = (const float*)d_in[9];
  const float* be2 = (const float*)d_in[10];
  const float* m2  = (const float*)d_in[11];
  const float* v2  = (const float*)d_in[12];
  const float* wa  = (const float*)d_in[13];
  const float* ba  = (const float*)d_in[14];
  const float* wd1 = (const float*)d_in[15];
  const float* bd1 = (const float*)d_in[16];
  const float* wd2 = (const float*)d_in[17];
  const float* bd2 = (const float*)d_in[18];

  const int B = 8192;
  float* f  = (float*)d_ws;                // B x 4096 flattened features
  float* h1 = f + (size_t)B * 4096;        // B x 512 hidden

  fused_conv<<<B, 256, 0, stream>>>(x, w1, b1, g1, be1, m1, v1,
                                    w2, b2, g2, be2, m2, v2, wa, ba, f);
  gemm_bias_act<0><<<dim3(512 / 32, B / 64), 256, 0, stream>>>(
      f, wd1, bd1, h1, B, 512, 4096);
  gemm_bias_act<1><<<dim3(128 / 32, B / 64), 256, 0, stream>>>(
      h1, wd2, bd2, (float*)d_out, B, 128, 512);
}